// GaussianKDE_39118562132363
// MI455X (gfx1250) — compile-verified
//
#include <hip/hip_runtime.h>
#include <math.h>

typedef __attribute__((ext_vector_type(2))) float v2f;
typedef __attribute__((ext_vector_type(8))) float v8f;

#define TPB_MAIN 256
#define WAVES_PER_BLOCK (TPB_MAIN / 32)
#define CHUNK 2048

// ---------------------------------------------------------------------------
// Kernel 1: block-wide reduction for sum(w), sum(w^2), sum(x), sum(x^2)
// ---------------------------------------------------------------------------
__global__ __launch_bounds__(1024)
void kde_reduce_kernel(const float* __restrict__ dataset,
                       const float* __restrict__ weights,
                       int n, float* __restrict__ sums) {
    __shared__ float sw[1024], sw2[1024], sx[1024], sx2[1024];
    int tid = threadIdx.x;
    float a = 0.f, b = 0.f, c = 0.f, d = 0.f;
    for (int i = tid; i < n; i += 1024) {
        float w = weights[i];
        float x = dataset[i];
        a += w; b += w * w; c += x; d += x * x;
    }
    sw[tid] = a; sw2[tid] = b; sx[tid] = c; sx2[tid] = d;
    __syncthreads();
    for (int s = 512; s > 0; s >>= 1) {
        if (tid < s) {
            sw[tid]  += sw[tid + s];
            sw2[tid] += sw2[tid + s];
            sx[tid]  += sx[tid + s];
            sx2[tid] += sx2[tid + s];
        }
        __syncthreads();
    }
    if (tid == 0) {
        sums[0] = sw[0]; sums[1] = sw2[0]; sums[2] = sx[0]; sums[3] = sx2[0];
    }
}

// ---------------------------------------------------------------------------
// Kernel 2: single-block bitonic sort of a +inf-padded copy (np = pow2 >= n)
// in global scratch. __syncthreads() orders global accesses within the block.
// ---------------------------------------------------------------------------
__global__ __launch_bounds__(1024)
void kde_sort_kernel(const float* __restrict__ dataset, int n, int np,
                     float* __restrict__ buf) {
    int tid = threadIdx.x;
    for (int i = tid; i < np; i += 1024)
        buf[i] = (i < n) ? dataset[i] : __builtin_inff();
    __syncthreads();
    for (int k = 2; k <= np; k <<= 1) {
        for (int j = k >> 1; j > 0; j >>= 1) {
            for (int i = tid; i < np; i += 1024) {
                int ixj = i ^ j;
                if (ixj > i) {
                    float x = buf[i], y = buf[ixj];
                    bool doswap = ((i & k) == 0) ? (x > y) : (x < y);
                    if (doswap) { buf[i] = y; buf[ixj] = x; }
                }
            }
            __syncthreads();
        }
    }
}

// ---------------------------------------------------------------------------
// Kernel 3: fold statistics into
//   r     = sqrt(0.5*log2(e)) / h   (distance pre-scale so the kernel entry
//                                    is exp2(-(r*xi - r*xj)^2))
//   alpha = 1/(h*sqrt(2*pi)*sum(w)) (weight normalization folded in)
// ---------------------------------------------------------------------------
__global__ void kde_scalars_kernel(const float* __restrict__ sums,
                                   const float* __restrict__ sorted,
                                   int n, float* __restrict__ coef) {
    if (threadIdx.x == 0 && blockIdx.x == 0) {
        float sumw = sums[0], sumw2 = sums[1], sumx = sums[2], sumx2 = sums[3];
        float fn = (float)n;
        float neff = sumw * sumw / sumw2;
        float var  = (sumx2 - sumx * sumx / fn) / (fn - 1.0f);  // ddof=1
        float sd   = sqrtf(var);
        float p25 = 0.25f * (fn - 1.0f);
        int   l25 = (int)p25;
        int   h25 = (l25 + 1 < n) ? l25 + 1 : n - 1;
        float f25 = p25 - (float)l25;
        float q25 = sorted[l25] + f25 * (sorted[h25] - sorted[l25]);
        float p75 = 0.75f * (fn - 1.0f);
        int   l75 = (int)p75;
        int   h75 = (l75 + 1 < n) ? l75 + 1 : n - 1;
        float f75 = p75 - (float)l75;
        float q75 = sorted[l75] + f75 * (sorted[h75] - sorted[l75]);
        float iqr = q75 - q25;
        float sig = fminf(sd, iqr / 1.34f);
        float h   = 0.9f * sig * powf(neff, -0.2f);
        // 0.5*log2(e) = 0.72134752044;  sqrt of that = 0.84932185419
        coef[0] = 0.84932185419f / h;
        coef[1] = 1.0f / (h * sqrtf(2.0f * (float)M_PI) * sumw);
    }
}

// ---------------------------------------------------------------------------
// Kernel 4: main KDE. One wave32 owns 16 output rows. For each group of 4
// columns j: A[m][k] = exp2(-(r*x_m - r*x_{j+k})^2) (fp32, 2 exps per lane,
// negate folds into the v_exp_f32 source modifier), B[k][n] = w_{j+k}
// replicated across columns. fp32 WMMA 16x16x4 accumulates
// D[m][*] = sum_k A[m][k]*w_k. Two accumulators break the D->C chain.
// A layout (ISA 7.12.2): lanes 0-15 -> M=lane, K=0(v0),1(v1);
//                        lanes 16-31 -> M=lane-16, K=2(v0),3(v1).
// B layout: lanes 0-15 -> K=0(v0),1(v1); lanes 16-31 -> K=2(v0),3(v1).
// D layout: VGPR r -> rows r (lanes 0-15) / r+8 (lanes 16-31).
// ---------------------------------------------------------------------------
__global__ __launch_bounds__(TPB_MAIN)
void kde_main_kernel(const float* __restrict__ dataset,
                     const float* __restrict__ weights,
                     int n, const float* __restrict__ coef,
                     float* __restrict__ out) {
    __shared__ float xs[CHUNK];   // holds r * x_j
    __shared__ float wsh[CHUNK];  // holds w_j

    const int lane = threadIdx.x & 31;
    const int wave = threadIdx.x >> 5;
    const int ntiles = (n + 15) >> 4;
    int tile = blockIdx.x * WAVES_PER_BLOCK + wave;
    if (tile >= ntiles) tile = ntiles - 1;   // redundant duplicate work; keeps EXEC all-ones

    const float r     = coef[0];
    const float alpha = coef[1];

    int row_m = tile * 16 + (lane & 15);
    float xir = r * dataset[min(row_m, n - 1)];
    const int khalf = (lane >> 4) << 1;      // 0 for lanes 0-15, 2 for lanes 16-31

    v8f acc0 = {};
    v8f acc1 = {};

    for (int j0 = 0; j0 < n; j0 += CHUNK) {
        __syncthreads();
        for (int t = threadIdx.x; t < CHUNK; t += TPB_MAIN) {
            int j = j0 + t;
            bool ok = (j < n);
            xs[t]  = ok ? r * dataset[j] : 0.0f;  // padded cols: w=0 -> contribution 0
            wsh[t] = ok ? weights[j] : 0.0f;
        }
        __syncthreads();

        for (int jj = 0; jj < CHUNK; jj += 8) {
            float2 xv0 = *(const float2*)&xs[jj + khalf];
            float2 wv0 = *(const float2*)&wsh[jj + khalf];
            float2 xv1 = *(const float2*)&xs[jj + 4 + khalf];
            float2 wv1 = *(const float2*)&wsh[jj + 4 + khalf];

            float d00 = xir - xv0.x, d01 = xir - xv0.y;
            float d10 = xir - xv1.x, d11 = xir - xv1.y;

            v2f a0, b0, a1, b1;
            a0[0] = __builtin_amdgcn_exp2f(-(d00 * d00));
            a0[1] = __builtin_amdgcn_exp2f(-(d01 * d01));
            a1[0] = __builtin_amdgcn_exp2f(-(d10 * d10));
            a1[1] = __builtin_amdgcn_exp2f(-(d11 * d11));
            b0[0] = wv0.x; b0[1] = wv0.y;
            b1[0] = wv1.x; b1[1] = wv1.y;

            acc0 = __builtin_amdgcn_wmma_f32_16x16x4_f32(
                false, a0, false, b0, (short)0, acc0, false, false);
            acc1 = __builtin_amdgcn_wmma_f32_16x16x4_f32(
                false, a1, false, b1, (short)0, acc1, false, false);
        }
    }

    v8f acc = acc0 + acc1;

    // Every column of D is identical; lane 0 writes rows 0-7, lane 16 rows 8-15.
    if (lane == 0 || lane == 16) {
        int mbase = tile * 16 + ((lane == 16) ? 8 : 0);
        for (int rr = 0; rr < 8; ++rr) {
            int row = mbase + rr;
            if (row < n) out[row] = logf(alpha * acc[rr] + 1e-10f);
        }
    }
}

// ---------------------------------------------------------------------------
extern "C" void kernel_launch(void* const* d_in, const int* in_sizes, int n_in,
                              void* d_out, int out_size, void* d_ws, size_t ws_size,
                              hipStream_t stream) {
    const float* dataset = (const float*)d_in[0];
    const float* weights = (const float*)d_in[1];
    float* out = (float*)d_out;
    int n = in_sizes[0];

    int np = 1;
    while (np < n) np <<= 1;           // padded pow2 size for bitonic sort

    float* ws      = (float*)d_ws;
    float* sortbuf = ws;               // np floats
    float* sums    = ws + np;          // 4 floats
    float* coef    = ws + np + 4;      // 2 floats

    kde_reduce_kernel<<<1, 1024, 0, stream>>>(dataset, weights, n, sums);
    kde_sort_kernel<<<1, 1024, 0, stream>>>(dataset, n, np, sortbuf);
    kde_scalars_kernel<<<1, 64, 0, stream>>>(sums, sortbuf, n, coef);

    int ntiles  = (n + 15) / 16;
    int nblocks = (ntiles + WAVES_PER_BLOCK - 1) / WAVES_PER_BLOCK;
    kde_main_kernel<<<nblocks, TPB_MAIN, 0, stream>>>(dataset, weights, n, coef, out);
}